// Model_80736795230564
// MI455X (gfx1250) — compile-verified
//
#include <hip/hip_runtime.h>
#include <math.h>

typedef __attribute__((ext_vector_type(2))) float v2f;
typedef __attribute__((ext_vector_type(8))) float v8f;

#define NEGINF (-1e30f)

__device__ __forceinline__ float warp_sum(float v) {
#pragma unroll
  for (int o = 16; o > 0; o >>= 1) v += __shfl_xor(v, o, 32);
  return v;
}

// blockDim.x must be 256
__device__ __forceinline__ void block_add256(float v, float* target) {
  __shared__ float sh[8];
  int lane = threadIdx.x & 31, wid = threadIdx.x >> 5;
  v = warp_sum(v);
  if (lane == 0) sh[wid] = v;
  __syncthreads();
  if (wid == 0) {
    float a = (lane < 8) ? sh[lane] : 0.f;
    a = warp_sum(a);
    if (lane == 0) atomicAdd(target, a);
  }
  __syncthreads();
}

// ---------------------------------------------------------------------------
// fp32 WMMA GEMM: C[M x N2] = act(A[M x K] @ W[N2 x K]^T + bias)
// One wave per 16(M) x 32(N) output tile: two v_wmma_f32_16x16x4_f32 per
// k-step sharing the A fragment. Branchless: out-of-range A/W rows are
// address-clamped (they only feed output rows/cols that the guarded store
// skips), so the inner loop is pure global_load_b64 + wmma, software
// pipelined one step ahead.
// A-frag (ISA 7.12.2, 32-bit A 16x4): lanes0-15 M=lane K={k0,k0+1},
// lanes16-31 M=lane-16 K={k0+2,k0+3}; B mirrors; C/D row = half*8+r, col=l16.
// ---------------------------------------------------------------------------
__global__ void k_gemm(const float* __restrict__ A, int lda,
                       const float* __restrict__ W, int ldw,
                       const float* __restrict__ bias,
                       float* __restrict__ C, int ldc,
                       int M, int N2, int K, long tiles, int tn2, int relu) {
  long tile = (long)blockIdx.x * (blockDim.x >> 5) + (threadIdx.x >> 5);
  if (tile >= tiles) return;                 // wave-uniform exit
  int tm_i = (int)(tile / tn2), tn_i = (int)(tile % tn2);
  int lane = threadIdx.x & 31;
  int half = lane >> 4, l16 = lane & 15;

  int arow = tm_i * 16 + l16;
  int wrow0 = tn_i * 32 + l16;
  int wrow1 = wrow0 + 16;
  // clamp (branchless safety; clamped rows only affect unstored outputs)
  int arc = min(arow, M - 1);
  int wr0 = min(wrow0, N2 - 1);
  int wr1 = min(wrow1, N2 - 1);
  const float* Ap  = A + (size_t)arc * lda + half * 2;
  const float* Wp0 = W + (size_t)wr0 * ldw + half * 2;
  const float* Wp1 = W + (size_t)wr1 * ldw + half * 2;

  v8f c0 = {0.f, 0.f, 0.f, 0.f, 0.f, 0.f, 0.f, 0.f};
  v8f c1 = c0;

  v2f a  = *(const v2f*)(Ap);
  v2f b0 = *(const v2f*)(Wp0);
  v2f b1 = *(const v2f*)(Wp1);
  for (int k0 = 4; k0 < K; k0 += 4) {
    v2f an  = *(const v2f*)(Ap + k0);
    v2f b0n = *(const v2f*)(Wp0 + k0);
    v2f b1n = *(const v2f*)(Wp1 + k0);
    c0 = __builtin_amdgcn_wmma_f32_16x16x4_f32(false, a, false, b0, (short)0, c0, false, false);
    c1 = __builtin_amdgcn_wmma_f32_16x16x4_f32(false, a, false, b1, (short)0, c1, false, false);
    a = an; b0 = b0n; b1 = b1n;
  }
  c0 = __builtin_amdgcn_wmma_f32_16x16x4_f32(false, a, false, b0, (short)0, c0, false, false);
  c1 = __builtin_amdgcn_wmma_f32_16x16x4_f32(false, a, false, b1, (short)0, c1, false, false);

  int col0 = tn_i * 32 + l16;
  int col1 = col0 + 16;
  float bv0 = (bias && col0 < N2) ? bias[col0] : 0.f;
  float bv1 = (bias && col1 < N2) ? bias[col1] : 0.f;
#pragma unroll
  for (int r = 0; r < 8; r++) {
    int row = tm_i * 16 + half * 8 + r;
    if (row < M) {
      if (col0 < N2) {
        float v = c0[r] + bv0;
        if (relu) v = fmaxf(v, 0.f);
        C[(size_t)row * ldc + col0] = v;
      }
      if (col1 < N2) {
        float v = c1[r] + bv1;
        if (relu) v = fmaxf(v, 0.f);
        C[(size_t)row * ldc + col1] = v;
      }
    }
  }
}

// ---------------------------------------------------------------------------
// small utility kernels
// ---------------------------------------------------------------------------
__global__ void k_setf(float* p, float v, long n) {
  long i = (long)blockIdx.x * blockDim.x + threadIdx.x;
  if (i < n) p[i] = v;
}
__global__ void k_seti(int* p, int v, long n) {
  long i = (long)blockIdx.x * blockDim.x + threadIdx.x;
  if (i < n) p[i] = v;
}

// degree with self loop (w=1), over doubled edges
__global__ void k_scatter_deg(const int* __restrict__ es, const int* __restrict__ ed,
                              const float* __restrict__ ew, float* deg, int E) {
  int t = blockIdx.x * blockDim.x + threadIdx.x;
  if (t >= 2 * E) return;
  int d = (t < E) ? ed[t] : es[t - E];
  float w = ew[t < E ? t : t - E];
  atomicAdd(&deg[d], w);
}
__global__ void k_dinv(const float* deg, float* dinv, int N) {
  int n = blockIdx.x * blockDim.x + threadIdx.x;
  if (n < N) dinv[n] = (deg[n] > 0.f) ? rsqrtf(deg[n]) : 0.f;
}

__global__ void k_build_x0(const float* __restrict__ ns, const int* __restrict__ nt,
                           float* __restrict__ hcat, int N) {
  int n = blockIdx.x * blockDim.x + threadIdx.x;
  if (n >= N) return;
  float* r = hcat + (size_t)n * 28;
  r[0] = ns[n * 3 + 0]; r[1] = ns[n * 3 + 1]; r[2] = ns[n * 3 + 2];
  int t = nt[n];
#pragma unroll
  for (int c = 0; c < 4; c++) r[3 + c] = (t == c) ? 1.f : 0.f;
}

// h = X(7) @ W(7x7)^T, X strided inside hcat
__global__ void k_gcn_xw(const float* __restrict__ X, int ldx,
                         const float* __restrict__ W, float* __restrict__ Ho, int N) {
  int n = blockIdx.x * blockDim.x + threadIdx.x;
  if (n >= N) return;
  const float* x = X + (size_t)n * ldx;
  float xv[7];
#pragma unroll
  for (int i = 0; i < 7; i++) xv[i] = x[i];
#pragma unroll
  for (int o = 0; o < 7; o++) {
    float s = 0.f;
#pragma unroll
    for (int i = 0; i < 7; i++) s += W[o * 7 + i] * xv[i];
    Ho[(size_t)n * 7 + o] = s;
  }
}
__global__ void k_gcn_self(const float* __restrict__ h, const float* __restrict__ dinv,
                           float* __restrict__ out, int N) {
  int n = blockIdx.x * blockDim.x + threadIdx.x;
  if (n >= N) return;
  float nn = dinv[n] * dinv[n];
#pragma unroll
  for (int o = 0; o < 7; o++) out[(size_t)n * 7 + o] = nn * h[(size_t)n * 7 + o];
}
__global__ void k_gcn_scatter(const float* __restrict__ h, const float* __restrict__ dinv,
                              const int* __restrict__ es, const int* __restrict__ ed,
                              const float* __restrict__ ew, float* __restrict__ out, int E) {
  int t = blockIdx.x * blockDim.x + threadIdx.x;
  if (t >= 2 * E) return;
  int s = (t < E) ? es[t] : ed[t - E];
  int d = (t < E) ? ed[t] : es[t - E];
  float w = ew[t < E ? t : t - E];
  float nrm = dinv[s] * w * dinv[d];
#pragma unroll
  for (int o = 0; o < 7; o++) atomicAdd(&out[(size_t)d * 7 + o], nrm * h[(size_t)s * 7 + o]);
}
__global__ void k_gcn_out(const float* __restrict__ acc7, const float* __restrict__ b,
                          float* __restrict__ hcat, int outoff, int N) {
  int n = blockIdx.x * blockDim.x + threadIdx.x;
  if (n >= N) return;
#pragma unroll
  for (int o = 0; o < 7; o++)
    hcat[(size_t)n * 28 + outoff + o] = fmaxf(acc7[(size_t)n * 7 + o] + b[o], 0.f);
}

// z = mu + eps*sigma ; accumulate 0.5*sum(log var + (eps*sg)^2/var) and 0.5*sum z^2
__global__ void k_z(const float* __restrict__ mu, const float* __restrict__ sg,
                    const float* __restrict__ eps, float* __restrict__ z,
                    float* acc, long total) {
  long i = (long)blockIdx.x * blockDim.x + threadIdx.x;
  float q = 0.f, zz = 0.f;
  if (i < total) {
    float m = mu[i], s = sg[i], e = eps[i];
    float zv = m + e * s;
    z[i] = zv;
    float var = fmaxf(s, 1e-6f);
    float d = e * s;
    q = 0.5f * (logf(var) + d * d / var);
    zz = 0.5f * zv * zv;
  }
  __shared__ float sh[2][8];
  int lane = threadIdx.x & 31, wid = threadIdx.x >> 5;
  q = warp_sum(q); zz = warp_sum(zz);
  if (lane == 0) { sh[0][wid] = q; sh[1][wid] = zz; }
  __syncthreads();
  if (wid == 0) {
    float a = (lane < 8) ? sh[0][lane] : 0.f;
    float b = (lane < 8) ? sh[1][lane] : 0.f;
    a = warp_sum(a); b = warp_sum(b);
    if (lane == 0) { atomicAdd(&acc[0], a); atomicAdd(&acc[1], b); }
  }
}

// t path: tbase = z @ t1_W[:, :H]^T already done; finish per node.
__global__ void k_tlogs(const float* __restrict__ tbase, const float* __restrict__ t1W,
                        const float* __restrict__ t1b, const float* __restrict__ t2W,
                        const float* __restrict__ t2b, const int* __restrict__ nt,
                        float* acc, int N, int H, int D260) {
  int n = blockIdx.x * blockDim.x + threadIdx.x;
  float part = 0.f;
  if (n < N) {
    const float* tb = tbase + (size_t)n * D260;
    float logs[4];
    for (int c = 0; c < 4; c++) {
      float s = t2b[0];
      for (int k = 0; k < D260; k++) {
        float h = tb[k] + t1b[k] + t1W[(size_t)k * D260 + H + c];
        s += t2W[k] * fmaxf(h, 0.f);
      }
      logs[c] = s;
    }
    float m = fmaxf(fmaxf(logs[0], logs[1]), fmaxf(logs[2], logs[3]));
    float se = expf(logs[0] - m) + expf(logs[1] - m) + expf(logs[2] - m) + expf(logs[3] - m);
    part = (m + logf(se)) - logs[nt[n] & 3];
  }
  block_add256(part, &acc[2]);
}

__global__ void k_transpose(const float* __restrict__ A, float* __restrict__ At, int H) {
  int i = blockIdx.x * blockDim.x + threadIdx.x;
  if (i >= H * H) return;
  int j = i / H, k = i % H;
  At[(size_t)k * H + j] = A[(size_t)j * H + k];
}

// sequential tanh RNN; one block of 256 threads; h kept in LDS, WhhT L2-resident.
__global__ void k_rnn(const float* __restrict__ u, const float* __restrict__ whhT,
                      const float* __restrict__ bhh, const float* __restrict__ lW,
                      const float* __restrict__ lb, float* acc, int N, int H, float Ef) {
  __shared__ float hsh[256];
  __shared__ float red[8];
  int j = threadIdx.x;
  hsh[j] = 0.f;
  __syncthreads();
  for (int t = 0; t < N; t++) {
    float s = u[(size_t)t * H + j] + bhh[j];
    for (int k = 0; k < 256; k += 4) {
      s += whhT[(size_t)(k + 0) * H + j] * hsh[k + 0];
      s += whhT[(size_t)(k + 1) * H + j] * hsh[k + 1];
      s += whhT[(size_t)(k + 2) * H + j] * hsh[k + 2];
      s += whhT[(size_t)(k + 3) * H + j] * hsh[k + 3];
    }
    __syncthreads();
    hsh[j] = tanhf(s);
    __syncthreads();
  }
  float p = hsh[j] * lW[j];
  p = warp_sum(p);
  if ((j & 31) == 0) red[j >> 5] = p;
  __syncthreads();
  if (j == 0) {
    float lv = lb[0];
    for (int w = 0; w < 8; w++) lv += red[w];
    acc[3] = expf(lv) - Ef * lv;
  }
}

// zp[k] = [z[cand_a[k]], z[cand_b[k]]]
__global__ void k_zp(const float* __restrict__ z, const int* __restrict__ es,
                     const int* __restrict__ ed, const int* __restrict__ nsx,
                     const int* __restrict__ ndx, float* __restrict__ zp,
                     int E, int K2, int H) {
  long i = (long)blockIdx.x * blockDim.x + threadIdx.x;
  long total = (long)K2 * 2 * H;
  if (i >= total) return;
  int k = (int)(i / (2 * H));
  int c = (int)(i % (2 * H));
  int isB = (c >= H);
  int col = isB ? c - H : c;
  int nd = (k < E) ? (isB ? ed[k] : es[k]) : (isB ? ndx[k - E] : nsx[k - E]);
  zp[i] = z[(size_t)nd * H + col];
}

__global__ void k_elogs(const float* __restrict__ ehid, const float* __restrict__ e2W,
                        const float* __restrict__ e2b, float* __restrict__ logits,
                        int K2, int E, int twoH, float logrf) {
  int k = blockIdx.x * blockDim.x + threadIdx.x;
  if (k >= K2) return;
  const float* h = ehid + (size_t)k * twoH;
  float s = e2b[0];
  for (int c = 0; c < twoH; c++) s += h[c] * e2W[c];
  logits[k] = s + ((k < E) ? 0.f : logrf);
}

__global__ void k_rlogs3(const float* __restrict__ rhid, const float* __restrict__ r2W,
                         const float* __restrict__ r2b, float* __restrict__ rlogs,
                         int E, int twoH) {
  int e = blockIdx.x * blockDim.x + threadIdx.x;
  if (e >= E) return;
  const float* h = rhid + (size_t)e * twoH;
#pragma unroll
  for (int r = 0; r < 3; r++) {
    float s = r2b[r];
    for (int c = 0; c < twoH; c++) s += h[c] * r2W[r * twoH + c];
    rlogs[e * 3 + r] = s;
  }
}

// per-node edge-touch lists for cap() queries
__global__ void k_cnt(const int* es, const int* ed, int* cnt, int E) {
  int e = blockIdx.x * blockDim.x + threadIdx.x;
  if (e >= E) return;
  atomicAdd(&cnt[es[e]], 1);
  atomicAdd(&cnt[ed[e]], 1);
}
__global__ void k_scan(const int* __restrict__ cnt, int* offs, int* fillpos, int N) {
  if (blockIdx.x == 0 && threadIdx.x == 0) {
    int a = 0;
    for (int n = 0; n < N; n++) { offs[n] = a; fillpos[n] = a; a += cnt[n]; }
    offs[N] = a;
  }
}
__global__ void k_fill(const int* es, const int* ed, const float* ew,
                       int* fillpos, int* leidx, float* lew, int E) {
  int e = blockIdx.x * blockDim.x + threadIdx.x;
  if (e >= E) return;
  int p = atomicAdd(&fillpos[es[e]], 1);
  leidx[p] = e; lew[p] = ew[e];
  p = atomicAdd(&fillpos[ed[e]], 1);
  leidx[p] = e; lew[p] = ew[e];
}

__device__ __forceinline__ float capf(int node, int i, const int* __restrict__ offs,
                                      const int* __restrict__ leidx,
                                      const float* __restrict__ lew,
                                      const int* __restrict__ nt) {
  const float val[4] = {4.f, 1.f, 6.f, 5.f};
  float c = val[nt[node] & 3];
  int b = offs[node], e = offs[node + 1];
  for (int p = b; p < e; p++)
    if (leidx[p] < i) c -= lew[p];
  return c;
}

// one wave per row i: online masked logsumexp over K2 candidates
__global__ void k_mlle(const float* __restrict__ logits, const int* __restrict__ es,
                       const int* __restrict__ ed, const int* __restrict__ nsx,
                       const int* __restrict__ ndx, const int* __restrict__ offs,
                       const int* __restrict__ leidx, const float* __restrict__ lew,
                       const int* __restrict__ nt, float* acc, int E, int K2) {
  int i = blockIdx.x;
  int lane = threadIdx.x;
  float m = NEGINF, s = 0.f;
  for (int k = lane; k < K2; k += 32) {
    if (k < i && k < E) continue;               // mask: (k>=i)|(k>=E)
    int a = (k < E) ? es[k] : nsx[k - E];
    int b = (k < E) ? ed[k] : ndx[k - E];
    float ca = capf(a, i, offs, leidx, lew, nt);
    float cb = capf(b, i, offs, leidx, lew, nt);
    if (fminf(ca, cb) <= 0.f) continue;         // mask: cap_min > 0
    float L = logits[k];
    if (L > m) { s = s * expf(m - L) + 1.f; m = L; }
    else s += expf(L - m);
  }
#pragma unroll
  for (int o = 16; o > 0; o >>= 1) {
    float m2 = __shfl_xor(m, o, 32), s2 = __shfl_xor(s, o, 32);
    float M = fmaxf(m, m2);
    s = s * expf(m - M) + s2 * expf(m2 - M);
    m = M;
  }
  if (lane == 0) {
    float lse = (s > 0.f) ? m + logf(s) : NEGINF;
    atomicAdd(&acc[4], lse - logits[i]);
  }
}

__global__ void k_mllr(const float* __restrict__ rlogs, const int* __restrict__ es,
                       const int* __restrict__ ed, const float* __restrict__ ew,
                       const int* __restrict__ offs, const int* __restrict__ leidx,
                       const float* __restrict__ lew, const int* __restrict__ nt,
                       float* acc, int E) {
  int i = blockIdx.x * blockDim.x + threadIdx.x;
  float part = 0.f;
  if (i < E) {
    float cape = fminf(capf(es[i], i, offs, leidx, lew, nt),
                       capf(ed[i], i, offs, leidx, lew, nt));
    float m = NEGINF, s = 0.f;
#pragma unroll
    for (int w = 0; w < 3; w++) {
      if ((float)(w + 1) <= cape) {
        float L = rlogs[i * 3 + w];
        if (L > m) { s = s * expf(m - L) + 1.f; m = L; }
        else s += expf(L - m);
      }
    }
    float lser = (s > 0.f) ? m + logf(s) : NEGINF;
    int widx = (int)ew[i] - 1;
    widx = min(max(widx, 0), 2);
    part = lser - rlogs[i * 3 + widx];
  }
  block_add256(part, &acc[5]);
}

// agg path scatters (H columns per directed edge, one block per edge)
__global__ void k_agg_self(const float* __restrict__ h2, const float* __restrict__ dinv,
                           float* __restrict__ agg, int N, int H) {
  long i = (long)blockIdx.x * blockDim.x + threadIdx.x;
  if (i >= (long)N * H) return;
  int n = (int)(i / H);
  agg[i] = dinv[n] * dinv[n] * h2[i];
}
__global__ void k_agg_scatter(const float* __restrict__ h2, const float* __restrict__ dinv,
                              const int* __restrict__ es, const int* __restrict__ ed,
                              const float* __restrict__ ew, float* __restrict__ agg,
                              int E, int H) {
  int t = blockIdx.x;
  int s = (t < E) ? es[t] : ed[t - E];
  int d = (t < E) ? ed[t] : es[t - E];
  float w = ew[t < E ? t : t - E];
  float nrm = dinv[s] * w * dinv[d];
  int c = threadIdx.x;
  atomicAdd(&agg[(size_t)d * H + c], nrm * h2[(size_t)s * H + c]);
}
__global__ void k_agg_fin(float* __restrict__ agg, const float* __restrict__ b,
                          int N, int H) {
  long i = (long)blockIdx.x * blockDim.x + threadIdx.x;
  if (i >= (long)N * H) return;
  int c = (int)(i % H);
  agg[i] = fmaxf(agg[i] + b[c], 0.f);
}

__global__ void k_mlls(const float* __restrict__ mhid, const float* __restrict__ shid,
                       const float* __restrict__ smu2W, const float* __restrict__ smu2b,
                       const float* __restrict__ ssg2W, const float* __restrict__ ssg2b,
                       const float* __restrict__ ns, float* acc, int N, int H) {
  int n = blockIdx.x * blockDim.x + threadIdx.x;
  float part = 0.f;
  if (n < N) {
    const float* mh = mhid + (size_t)n * H;
    const float* sh2 = shid + (size_t)n * H;
    float d[3], v[3];
#pragma unroll
    for (int r = 0; r < 3; r++) {
      float s1 = smu2b[r], s2 = ssg2b[r];
      for (int c = 0; c < H; c++) { s1 += smu2W[r * H + c] * mh[c]; s2 += ssg2W[r * H + c] * sh2[c]; }
      d[r] = ns[n * 3 + r] - s1;
      v[r] = s2;
    }
    float vv = v[0] * v[0] + v[1] * v[1] + v[2] * v[2];
    float dv = d[0] * v[0] + d[1] * v[1] + d[2] * v[2];
    float dd = d[0] * d[0] + d[1] * d[1] + d[2] * d[2];
    const float EPS = 1e-4f;
    float quad = (dd - dv * dv / (EPS + vv)) / EPS;
    float logdet = 3.f * logf(EPS) + log1pf(vv / EPS);
    part = 0.5f * (quad + logdet + 3.f * logf(2.f * 3.14159265358979323846f));
  }
  block_add256(part, &acc[6]);
}

__global__ void k_final(const float* acc, const float* lamp, float* out, int N) {
  if (blockIdx.x == 0 && threadIdx.x == 0) {
    float lam = lamp[0];
    float mlln = lam - (float)N * logf(lam + 1e-8f);
    out[0] = mlln + acc[1] + acc[2] + acc[3] + acc[4] + acc[5] + acc[6] - acc[0];
  }
}

// ---------------------------------------------------------------------------
// host
// ---------------------------------------------------------------------------
static inline void gemm(hipStream_t st, const float* A, int lda, const float* W, int ldw,
                        const float* bias, float* C, int ldc, int M, int N2, int K, int relu) {
  int tm = (M + 15) / 16, tn2 = (N2 + 31) / 32;
  long tiles = (long)tm * tn2;
  const int WPB = 4;  // waves per block
  unsigned blocks = (unsigned)((tiles + WPB - 1) / WPB);
  k_gemm<<<dim3(blocks), dim3(32 * WPB), 0, st>>>(A, lda, W, ldw, bias, C, ldc,
                                                  M, N2, K, tiles, tn2, relu);
}

extern "C" void kernel_launch(void* const* d_in, const int* in_sizes, int n_in,
                              void* d_out, int out_size, void* d_ws, size_t ws_size,
                              hipStream_t stream) {
  const float* node_s = (const float*)d_in[0];
  const int*   node_t = (const int*)d_in[1];
  const int*   esrc   = (const int*)d_in[2];
  const int*   edst   = (const int*)d_in[3];
  const float* ew     = (const float*)d_in[4];
  const int*   nsx    = (const int*)d_in[5];
  const int*   ndx    = (const int*)d_in[6];
  const float* eps    = (const float*)d_in[7];
  const float* lam    = (const float*)d_in[8];
  const float* gcnW   = (const float*)d_in[9];
  const float* gcnB   = (const float*)d_in[10];
  const float* mu1W = (const float*)d_in[11]; const float* mu1b = (const float*)d_in[12];
  const float* mu2W = (const float*)d_in[13]; const float* mu2b = (const float*)d_in[14];
  const float* sg1W = (const float*)d_in[15]; const float* sg1b = (const float*)d_in[16];
  const float* sg2W = (const float*)d_in[17]; const float* sg2b = (const float*)d_in[18];
  const float* t1W  = (const float*)d_in[19]; const float* t1b  = (const float*)d_in[20];
  const float* t2W  = (const float*)d_in[21]; const float* t2b  = (const float*)d_in[22];
  const float* Wih  = (const float*)d_in[23]; const float* Whh  = (const float*)d_in[24];
  const float* bih  = (const float*)d_in[25]; const float* bhh  = (const float*)d_in[26];
  const float* lW   = (const float*)d_in[27]; const float* lb   = (const float*)d_in[28];
  const float* e1W  = (const float*)d_in[29]; const float* e1b  = (const float*)d_in[30];
  const float* e2W  = (const float*)d_in[31]; const float* e2b  = (const float*)d_in[32];
  const float* r1W  = (const float*)d_in[33]; const float* r1b  = (const float*)d_in[34];
  const float* r2W  = (const float*)d_in[35]; const float* r2b  = (const float*)d_in[36];
  const float* gsW  = (const float*)d_in[37]; const float* gsb  = (const float*)d_in[38];
  const float* smu1W = (const float*)d_in[39]; const float* smu1b = (const float*)d_in[40];
  const float* smu2W = (const float*)d_in[41]; const float* smu2b = (const float*)d_in[42];
  const float* ssg1W = (const float*)d_in[43]; const float* ssg1b = (const float*)d_in[44];
  const float* ssg2W = (const float*)d_in[45]; const float* ssg2b = (const float*)d_in[46];

  const int N = in_sizes[1];
  const int E = in_sizes[2];
  const int NEGn = in_sizes[5];
  const int H = in_sizes[7] / N;       // 256
  const int K2 = E + NEGn;
  const int D260 = H + 4;
  const int twoH = 2 * H;

  double P = (double)N * (double)(N - 1) * 0.5;
  double rf = (P - E) + (P - E) / (double)NEGn;
  float logrf = (float)log(rf);

  // workspace layout (floats); every block rounded to even so v2f loads stay
  // 8-byte aligned
  float* w = (float*)d_ws;
  size_t off = 0;
  auto F = [&](size_t n) { float* p = w + off; off += (n + 1) & ~(size_t)1; return p; };
  float* acc   = F(16);
  float* deg   = F(N);
  float* dinv  = F(N);
  float* hcat  = F((size_t)N * 28);
  float* tmpA  = F((size_t)N * 7);
  float* tmpB  = F((size_t)N * 7);
  float* hid28 = F((size_t)N * 28);
  float* encmu = F((size_t)N * H);
  float* encsg = F((size_t)N * H);
  float* z     = F((size_t)N * H);
  float* tbase = F((size_t)N * D260);
  float* u     = F((size_t)N * H);
  float* whhT  = F((size_t)H * H);
  float* zp    = F((size_t)K2 * twoH);
  float* ehid  = F((size_t)K2 * twoH);
  float* rhid  = F((size_t)E * twoH);
  float* logits = F(K2);
  float* rlogs  = F((size_t)E * 3);
  int* cnt     = (int*)F(N);
  int* offs    = (int*)F(N + 1);
  int* fillpos = (int*)F(N);
  int* leidx   = (int*)F((size_t)2 * E);
  float* lew   = F((size_t)2 * E);
  float* h2    = F((size_t)N * H);
  float* agg   = F((size_t)N * H);
  float* mhid  = F((size_t)N * H);
  float* shid  = F((size_t)N * H);
  (void)ws_size; (void)n_in; (void)out_size;

  auto g1 = [](long n) { return dim3((unsigned)((n + 255) / 256)); };
  dim3 B256(256);

  // accumulators
  k_setf<<<dim3(1), dim3(16), 0, stream>>>(acc, 0.f, 16);

  // degrees / norm
  k_setf<<<g1(N), B256, 0, stream>>>(deg, 1.f, N);     // self-loop weight
  k_scatter_deg<<<g1(2 * E), B256, 0, stream>>>(esrc, edst, ew, deg, E);
  k_dinv<<<g1(N), B256, 0, stream>>>(deg, dinv, N);

  // GCN stack -> hcat
  k_build_x0<<<g1(N), B256, 0, stream>>>(node_s, node_t, hcat, N);
  for (int l = 0; l < 3; l++) {
    k_gcn_xw<<<g1(N), B256, 0, stream>>>(hcat + l * 7, 28, gcnW + l * 49, tmpA, N);
    k_gcn_self<<<g1(N), B256, 0, stream>>>(tmpA, dinv, tmpB, N);
    k_gcn_scatter<<<g1(2 * E), B256, 0, stream>>>(tmpA, dinv, esrc, edst, ew, tmpB, E);
    k_gcn_out<<<g1(N), B256, 0, stream>>>(tmpB, gcnB + l * 7, hcat, (l + 1) * 7, N);
  }

  // encoder (WMMA GEMMs)
  gemm(stream, hcat, 28, mu1W, 28, mu1b, hid28, 28, N, 28, 28, 1);
  gemm(stream, hid28, 28, mu2W, 28, mu2b, encmu, H, N, H, 28, 0);
  gemm(stream, hcat, 28, sg1W, 28, sg1b, hid28, 28, N, 28, 28, 1);
  gemm(stream, hid28, 28, sg2W, 28, sg2b, encsg, H, N, H, 28, 1);
  k_z<<<g1((long)N * H), B256, 0, stream>>>(encmu, encsg, eps, z, acc, (long)N * H);

  // type head: tbase = z @ t1W[:, :H]^T
  gemm(stream, z, H, t1W, D260, nullptr, tbase, D260, N, D260, H, 0);
  k_tlogs<<<g1(N), B256, 0, stream>>>(tbase, t1W, t1b, t2W, t2b, node_t, acc, N, H, D260);

  // RNN
  gemm(stream, z, H, Wih, H, bih, u, H, N, H, H, 0);
  k_transpose<<<g1((long)H * H), B256, 0, stream>>>(Whh, whhT, H);
  k_rnn<<<dim3(1), B256, 0, stream>>>(u, whhT, bhh, lW, lb, acc, N, H, (float)E);

  // edge existence head
  k_zp<<<g1((long)K2 * twoH), B256, 0, stream>>>(z, esrc, edst, nsx, ndx, zp, E, K2, H);
  gemm(stream, zp, twoH, e1W, twoH, e1b, ehid, twoH, K2, twoH, twoH, 1);
  k_elogs<<<g1(K2), B256, 0, stream>>>(ehid, e2W, e2b, logits, K2, E, twoH, logrf);

  // edge weight head
  gemm(stream, zp, twoH, r1W, twoH, r1b, rhid, twoH, E, twoH, twoH, 1);
  k_rlogs3<<<g1(E), B256, 0, stream>>>(rhid, r2W, r2b, rlogs, E, twoH);

  // per-node touch lists (sparse cumsum(delta) replacement)
  k_seti<<<g1(N), B256, 0, stream>>>(cnt, 0, N);
  k_cnt<<<g1(E), B256, 0, stream>>>(esrc, edst, cnt, E);
  k_scan<<<dim3(1), dim3(32), 0, stream>>>(cnt, offs, fillpos, N);
  k_fill<<<g1(E), B256, 0, stream>>>(esrc, edst, ew, fillpos, leidx, lew, E);

  // masked logsumexp losses
  k_mlle<<<dim3((unsigned)E), dim3(32), 0, stream>>>(logits, esrc, edst, nsx, ndx,
                                                     offs, leidx, lew, node_t, acc, E, K2);
  k_mllr<<<g1(E), B256, 0, stream>>>(rlogs, esrc, edst, ew, offs, leidx, lew, node_t, acc, E);

  // structural head: agg = relu(gcn(z))
  gemm(stream, z, H, gsW, H, nullptr, h2, H, N, H, H, 0);
  k_agg_self<<<g1((long)N * H), B256, 0, stream>>>(h2, dinv, agg, N, H);
  k_agg_scatter<<<dim3((unsigned)(2 * E)), dim3((unsigned)H), 0, stream>>>(h2, dinv, esrc, edst, ew, agg, E, H);
  k_agg_fin<<<g1((long)N * H), B256, 0, stream>>>(agg, gsb, N, H);

  gemm(stream, agg, H, smu1W, H, smu1b, mhid, H, N, H, H, 1);
  gemm(stream, agg, H, ssg1W, H, ssg1b, shid, H, N, H, H, 1);
  k_mlls<<<g1(N), B256, 0, stream>>>(mhid, shid, smu2W, smu2b, ssg2W, ssg2b,
                                     node_s, acc, N, H);

  k_final<<<dim3(1), dim3(32), 0, stream>>>(acc, lam, (float*)d_out, N);
}